// MeanAggregator_76905684402638
// MI455X (gfx1250) — compile-verified
//
#include <hip/hip_runtime.h>
#include <math.h>
#include <stdint.h>

// Problem constants (from reference): N=100000, B=8192, K1=33, H=4, D=128
#define NB    8192
#define K1    33
#define NH    4
#define DD    128
#define TSTR  132   // padded LDS row stride in floats: 132%64==4 -> conflict-free b64 frag loads

typedef __attribute__((ext_vector_type(2))) float v2f;
typedef __attribute__((ext_vector_type(8))) float v8f;

// One wave32 per (b, h). Gathers the 33x128 neighbor tile to LDS with
// GLOBAL_LOAD_ASYNC_TO_LDS_B128 (ASYNCcnt path), computes attention scores
// with V_WMMA_F32_16X16X4_F32, softmax, weighted sum, writes out[b,h,0:128].
__global__ __launch_bounds__(32)
void gat_attn_kernel(const float* __restrict__ feat,
                     const float* __restrict__ a,
                     const long long* __restrict__ nidx,
                     float* __restrict__ out) {
    __shared__ float tile[K1 * TSTR];   // neighbor tile for this (b,h): 33 rows x 128 (stride 132)
    __shared__ float a_lds[2 * DD];     // a_n[h] (0..127), a_s[h] (128..255)
    __shared__ float sc[K1];            // leaky-relu'd scores

    const int wg   = blockIdx.x;        // 0 .. B*H-1
    const int b    = wg >> 2;
    const int h    = wg & 3;
    const int lane = threadIdx.x;       // 0..31
    const int half = lane >> 4;         // 0: lanes 0-15, 1: lanes 16-31
    const int lmod = lane & 15;

    // ---- async gather: 33 neighbor rows -> LDS (no VGPR destage, ASYNCcnt) ----
    const long long* idxp = nidx + (long long)b * K1;
    #pragma unroll 4
    for (int k = 0; k < K1; ++k) {
        long long n = idxp[k];  // uniform across wave -> scalarized load
        const float* src = feat + (long long)n * (NH * DD) + h * DD + 4 * lane;
        unsigned long long gaddr = (unsigned long long)(uintptr_t)src;
        unsigned int ldsoff = (unsigned int)(uintptr_t)(tile + k * TSTR + 4 * lane);
        asm volatile("global_load_async_to_lds_b128 %0, %1, off"
                     :: "v"(ldsoff), "v"(gaddr)
                     : "memory");
    }

    // ---- stage a[h] (256 floats) into LDS (overlaps with async gather) ----
    {
        const float4* ap  = (const float4*)(a + (size_t)h * 2 * DD);
        float4*       dst = (float4*)a_lds;
        dst[lane]      = ap[lane];
        dst[lane + 32] = ap[lane + 32];
    }

    // wait for the async tile before touching it
    asm volatile("s_wait_asynccnt 0" ::: "memory");

    // ---- self-row dots: dn = <neigh32, a_n>, ds = <neigh32, a_s> (butterfly reduce) ----
    float4 selfv = *(const float4*)(tile + (K1 - 1) * TSTR + 4 * lane);  // row 32 == "self"
    float4 an4   = *(const float4*)(a_lds + 4 * lane);
    float4 as4   = *(const float4*)(a_lds + DD + 4 * lane);
    float dn = selfv.x * an4.x + selfv.y * an4.y + selfv.z * an4.z + selfv.w * an4.w;
    float ds = selfv.x * as4.x + selfv.y * as4.y + selfv.z * as4.z + selfv.w * as4.w;
    #pragma unroll
    for (int m = 16; m >= 1; m >>= 1) {
        dn += __shfl_xor(dn, m, 32);
        ds += __shfl_xor(ds, m, 32);
    }

    // ---- WMMA scores for neighbor tiles k0=0 and k0=16 ----
    // D = A(16x4) x B(4x16) + C, accumulated over 32 chunks of K=4 (D=128).
    // A[m][kk] = a_n[4c+kk] (same for all m)   -> lane supplies a_n[4c+2*half +0/+1]
    // B[kk][n] = neigh[k0+n][4c+kk]            -> lane supplies tile[(k0+lmod)][4c+2*half +0/+1]
    // All D rows identical => every lane's C[0] = score[k0 + lmod].
    v8f c0 = {};
    v8f c1 = {};
    const int colbase = 2 * half;
    #pragma unroll 8
    for (int c = 0; c < DD / 4; ++c) {
        const int col = 4 * c + colbase;
        v2f afrag = *(const v2f*)(a_lds + col);                    // broadcast within half
        v2f bf0   = *(const v2f*)(tile + (0  + lmod) * TSTR + col);
        v2f bf1   = *(const v2f*)(tile + (16 + lmod) * TSTR + col);
        c0 = __builtin_amdgcn_wmma_f32_16x16x4_f32(false, afrag, false, bf0,
                                                   (short)0, c0, false, false);
        c1 = __builtin_amdgcn_wmma_f32_16x16x4_f32(false, afrag, false, bf1,
                                                   (short)0, c1, false, false);
    }

    // ---- assemble scores, leaky_relu(0.01), write to LDS ----
    // lane L<16 holds score[L] (tile0); lane L>=16 holds score[L] (=16+lmod, tile1)
    float s = (half ? c1[0] : c0[0]) + ds;
    s = (s > 0.f) ? s : 0.01f * s;
    sc[lane] = s;
    if (lane == 0) {
        float s32 = dn + ds;
        sc[32] = (s32 > 0.f) ? s32 : 0.01f * s32;
    }
    __syncthreads();   // single-wave WG: degenerates to NOP; LDS ops in-order per wave

    // ---- softmax over k (all lanes compute identical scalars from LDS broadcasts) ----
    float mx = -3.402823466e38f;
    #pragma unroll
    for (int k = 0; k < K1; ++k) mx = fmaxf(mx, sc[k]);
    float sum = 0.f;
    #pragma unroll
    for (int k = 0; k < K1; ++k) sum += __expf(sc[k] - mx);
    const float inv = 1.f / sum;

    // ---- weighted sum: out[b,h,d] = sum_k att[k] * tile[k][d] ----
    float4 acc = make_float4(0.f, 0.f, 0.f, 0.f);
    #pragma unroll 4
    for (int k = 0; k < K1; ++k) {
        float w = __expf(sc[k] - mx) * inv;
        float4 t = *(const float4*)(tile + k * TSTR + 4 * lane);
        acc.x += w * t.x;
        acc.y += w * t.y;
        acc.z += w * t.z;
        acc.w += w * t.w;
    }
    float4* op = (float4*)(out + ((long long)b * NH + h) * DD);
    op[lane] = acc;
}

extern "C" void kernel_launch(void* const* d_in, const int* in_sizes, int n_in,
                              void* d_out, int out_size, void* d_ws, size_t ws_size,
                              hipStream_t stream) {
    (void)in_sizes; (void)n_in; (void)out_size; (void)d_ws; (void)ws_size;
    const float*     feat = (const float*)d_in[0];
    const float*     a    = (const float*)d_in[1];
    const long long* nidx = (const long long*)d_in[2];
    float*           out  = (float*)d_out;

    dim3 grid(NB * NH);   // one wave32 workgroup per (b, h)
    dim3 block(32);
    gat_attn_kernel<<<grid, block, 0, stream>>>(feat, a, nidx, out);
}